// PolynomialLayer_46866683134251
// MI455X (gfx1250) — compile-verified
//
#include <hip/hip_runtime.h>
#include <hip/hip_bf16.h>

// PolynomialLayer: out[i] = [1, x0, x1, x2, x0^2, x0*x1, x0*x2, x1^2, x1*x2, x2^2]
// Pure streaming op: 416 MB of traffic, 48 MFLOP -> memory-bound ~18us floor on MI455X.
// Strategy: 4 rows per thread so both input (12 floats) and output (40 floats) are
// exact multiples of 128b -> all VMEM is GLOBAL_LOAD_B128 / GLOBAL_STORE_B128.
// Output stores are non-temporal (CDNA5 TH=NT) to avoid rinsing L2 with a
// write-once 320 MB stream.

typedef float v4f __attribute__((ext_vector_type(4)));

#define THREADS 256

__global__ __launch_bounds__(THREADS) void poly_features_kernel(
    const float* __restrict__ in, float* __restrict__ out, int n_rows) {
  const int n4 = n_rows >> 2;  // number of 4-row groups
  const int g = blockIdx.x * blockDim.x + threadIdx.x;

  if (g < n4) {
    const v4f* __restrict__ in4 = reinterpret_cast<const v4f*>(in) + (size_t)g * 3;
    // 4 rows = 12 floats = 3 x b128 loads (16B aligned: stride 48B)
    const v4f v0 = in4[0];  // {x0a, x1a, x2a, x0b}
    const v4f v1 = in4[1];  // {x1b, x2b, x0c, x1c}
    const v4f v2 = in4[2];  // {x2c, x0d, x1d, x2d}

    const float x0a = v0.x, x1a = v0.y, x2a = v0.z;
    const float x0b = v0.w, x1b = v1.x, x2b = v1.y;
    const float x0c = v1.z, x1c = v1.w, x2c = v2.x;
    const float x0d = v2.y, x1d = v2.z, x2d = v2.w;

    // 40 output floats = 10 x b128 NT stores (16B aligned: stride 160B).
    // Row r occupies out[r*10 .. r*10+9]; pack across row boundaries.
    v4f* __restrict__ out4 = reinterpret_cast<v4f*>(out) + (size_t)g * 10;

    v4f s0 = {1.0f, x0a, x1a, x2a};
    v4f s1 = {x0a * x0a, x0a * x1a, x0a * x2a, x1a * x1a};
    v4f s2 = {x1a * x2a, x2a * x2a, 1.0f, x0b};
    v4f s3 = {x1b, x2b, x0b * x0b, x0b * x1b};
    v4f s4 = {x0b * x2b, x1b * x1b, x1b * x2b, x2b * x2b};
    v4f s5 = {1.0f, x0c, x1c, x2c};
    v4f s6 = {x0c * x0c, x0c * x1c, x0c * x2c, x1c * x1c};
    v4f s7 = {x1c * x2c, x2c * x2c, 1.0f, x0d};
    v4f s8 = {x1d, x2d, x0d * x0d, x0d * x1d};
    v4f s9 = {x0d * x2d, x1d * x1d, x1d * x2d, x2d * x2d};

    __builtin_nontemporal_store(s0, out4 + 0);
    __builtin_nontemporal_store(s1, out4 + 1);
    __builtin_nontemporal_store(s2, out4 + 2);
    __builtin_nontemporal_store(s3, out4 + 3);
    __builtin_nontemporal_store(s4, out4 + 4);
    __builtin_nontemporal_store(s5, out4 + 5);
    __builtin_nontemporal_store(s6, out4 + 6);
    __builtin_nontemporal_store(s7, out4 + 7);
    __builtin_nontemporal_store(s8, out4 + 8);
    __builtin_nontemporal_store(s9, out4 + 9);
  }

  // Tail rows (n_rows % 4 <= 3), handled scalar by the first few threads.
  const int trow = (n4 << 2) + g;
  if (g < 4 && trow < n_rows) {
    const float x0 = in[(size_t)trow * 3 + 0];
    const float x1 = in[(size_t)trow * 3 + 1];
    const float x2 = in[(size_t)trow * 3 + 2];
    float* o = out + (size_t)trow * 10;
    o[0] = 1.0f; o[1] = x0; o[2] = x1; o[3] = x2;
    o[4] = x0 * x0; o[5] = x0 * x1; o[6] = x0 * x2;
    o[7] = x1 * x1; o[8] = x1 * x2; o[9] = x2 * x2;
  }
}

extern "C" void kernel_launch(void* const* d_in, const int* in_sizes, int n_in,
                              void* d_out, int out_size, void* d_ws, size_t ws_size,
                              hipStream_t stream) {
  (void)n_in; (void)out_size; (void)d_ws; (void)ws_size;
  const float* in = (const float*)d_in[0];
  float* out = (float*)d_out;
  const int n_rows = in_sizes[0] / 3;
  const int n4 = n_rows >> 2;
  int blocks = (n4 + THREADS - 1) / THREADS;
  if (blocks < 1) blocks = 1;  // ensure tail threads run even for tiny N
  poly_features_kernel<<<blocks, THREADS, 0, stream>>>(in, out, n_rows);
}